// LocLoss_1571958030912
// MI455X (gfx1250) — compile-verified
//
#include <hip/hip_runtime.h>
#include <hip/hip_bf16.h>
#include <math.h>

#define EPSF 1e-6f
#define BATCH 32
#define EMIT  128
#define K_PIX 16384          // 128*128
#define NTILES (K_PIX / 16)  // 1024 tiles of 16 pixels
#define WAVES 4
#define SHIFTF 16.0f
#define TWO_PI_P4 1558.5454565440132f  // (2*pi)^4
#define TILE_FLOATS 192                // 3 groups * 32 lanes * 2 floats

typedef __attribute__((ext_vector_type(2))) float v2f;
typedef __attribute__((ext_vector_type(8))) float v8f;

__device__ __forceinline__ float fast_rcp(float x) { return __builtin_amdgcn_rcpf(x); }

// ---------------------------------------------------------------------------
// Kernel 1: per-batch sum of clipped detection probabilities (mixture denom)
// ---------------------------------------------------------------------------
__global__ void k_sum_p(const float* __restrict__ p, float* __restrict__ sums) {
    __shared__ float red[256];
    const int b = blockIdx.x;
    float s = 0.0f;
    for (int k = threadIdx.x; k < K_PIX; k += 256) {
        float pv = p[(size_t)b * K_PIX + k];
        pv = fminf(fmaxf(pv, EPSF), 1.0f - EPSF);
        s += pv;
    }
    red[threadIdx.x] = s;
    __syncthreads();
    for (int o = 128; o > 0; o >>= 1) {
        if (threadIdx.x < o) red[threadIdx.x] += red[threadIdx.x + o];
        __syncthreads();
    }
    if (threadIdx.x == 0) sums[b] = red[0];
}

// ---------------------------------------------------------------------------
// Per-pixel feature computation (shared by both paths).
// B rows: 0..3 = i_c ; 4..7 = mu_c*i_c ; 8 = base - 0.5*sum(mu^2 i) + SHIFT
// ---------------------------------------------------------------------------
__device__ __forceinline__ void pixel_features(
    size_t idx, float inv_denom,
    const float* __restrict__ p,   const float* __restrict__ dxp,
    const float* __restrict__ dyp, const float* __restrict__ dzp,
    const float* __restrict__ php, const float* __restrict__ sxp,
    const float* __restrict__ syp, const float* __restrict__ szp,
    const float* __restrict__ spp,
    const float* __restrict__ pcx, const float* __restrict__ pcy,
    const float* __restrict__ pcz,
    float& ix, float& iy, float& iz, float& ip,
    float& mux, float& muy, float& muz, float& mup, float& c9)
{
    float pv  = fminf(fmaxf(p[idx], EPSF), 1.0f - EPSF);
    float lw  = __logf(pv * inv_denom + EPSF);
    mux = pcx[idx] + dxp[idx];
    muy = pcy[idx] + dyp[idx];
    muz = pcz[idx] + dzp[idx];
    mup = php[idx];
    float sx = fmaxf(sxp[idx], EPSF);
    float sy = fmaxf(syp[idx], EPSF);
    float sz = fmaxf(szp[idx], EPSF);
    float sp = fmaxf(spp[idx], EPSF);
    ix = fast_rcp(sx * sx + EPSF);
    iy = fast_rcp(sy * sy + EPSF);
    iz = fast_rcp(sz * sz + EPSF);
    ip = fast_rcp(sp * sp + EPSF);
    float log_norm = -0.5f * __logf(TWO_PI_P4 * (sx*sx) * (sy*sy) * (sz*sz) * (sp*sp));
    c9 = lw + log_norm
       - 0.5f * (mux*mux*ix + muy*muy*iy + muz*muz*iz + mup*mup*ip)
       + SHIFTF;
}

// ---------------------------------------------------------------------------
// Kernel 2a: precompute per-pixel B-matrix fragments, pre-swizzled into the
// WMMA lane layout so the main loop loads each fragment with one b64 load.
// feat[b][tile][g][lane][2]: g0 = inv-vars, g1 = mu*inv, g2 = bias row.
// ---------------------------------------------------------------------------
__global__ void k_features(
    const float* __restrict__ p,   const float* __restrict__ dxp,
    const float* __restrict__ dyp, const float* __restrict__ dzp,
    const float* __restrict__ php, const float* __restrict__ sxp,
    const float* __restrict__ syp, const float* __restrict__ szp,
    const float* __restrict__ spp,
    const float* __restrict__ pcx, const float* __restrict__ pcy,
    const float* __restrict__ pcz, const float* __restrict__ sum_p,
    float* __restrict__ feat)
{
    const size_t gid = (size_t)blockIdx.x * 256 + threadIdx.x;
    if (gid >= (size_t)BATCH * K_PIX) return;
    const int b    = (int)(gid >> 14);
    const int k    = (int)(gid & (K_PIX - 1));
    const int tile = k >> 4;
    const int n    = k & 15;

    const float inv_denom = fast_rcp(sum_p[b] + EPSF);
    float ix, iy, iz, ip, mux, muy, muz, mup, c9;
    pixel_features(gid, inv_denom, p, dxp, dyp, dzp, php, sxp, syp, szp, spp,
                   pcx, pcy, pcz, ix, iy, iz, ip, mux, muy, muz, mup, c9);

    float* tb = feat + ((size_t)b * NTILES + tile) * TILE_FLOATS;
    // group 0 rows {0,1 | 2,3}
    ((v2f*)(tb + 0 * 64))[n]      = (v2f){ix, iy};
    ((v2f*)(tb + 0 * 64))[16 + n] = (v2f){iz, ip};
    // group 1 rows {4,5 | 6,7}
    ((v2f*)(tb + 1 * 64))[n]      = (v2f){mux * ix, muy * iy};
    ((v2f*)(tb + 1 * 64))[16 + n] = (v2f){muz * iz, mup * ip};
    // group 2 rows {8,9 | 10,11}
    ((v2f*)(tb + 2 * 64))[n]      = (v2f){c9, 0.0f};
    ((v2f*)(tb + 2 * 64))[16 + n] = (v2f){0.0f, 0.0f};
}

// ---------------------------------------------------------------------------
// Kernel 2b: WMMA mixture logits + streaming exp-sum per emitter.
// grid = 32 batches * 8 emitter-tiles; block = 128 threads (4 wave32).
// USE_FEAT: load pre-swizzled fragments (3 x b64) ; else recompute in-loop.
// ---------------------------------------------------------------------------
template <bool USE_FEAT>
__global__ __launch_bounds__(128) void k_mixture(
    const float* __restrict__ p,   const float* __restrict__ dxp,
    const float* __restrict__ dyp, const float* __restrict__ dzp,
    const float* __restrict__ php, const float* __restrict__ sxp,
    const float* __restrict__ syp, const float* __restrict__ szp,
    const float* __restrict__ spp, const float* __restrict__ gt,
    const float* __restrict__ pcx, const float* __restrict__ pcy,
    const float* __restrict__ pcz, const float* __restrict__ sum_p,
    const float* __restrict__ feat, float* __restrict__ partials)
{
    __shared__ float red[WAVES][16];
    __shared__ float lsebuf[16];

    const int b    = blockIdx.x >> 3;
    const int et   = blockIdx.x & 7;
    const int wave = threadIdx.x >> 5;
    const int lane = threadIdx.x & 31;
    const int m    = lane & 15;
    const int h    = lane >> 4;

    // A fragments: A[e][0..3] = -0.5*g_c^2 ; [4..7] = g_c ; [8] = 1 ; pad 0
    const float* g = gt + ((size_t)b * EMIT + (size_t)et * 16 + m) * 4;
    const float gx = g[0], gy = g[1], gz = g[2], gp = g[3];
    v2f a0, a1, a2;
    if (h == 0) {
        a0 = (v2f){-0.5f * gx * gx, -0.5f * gy * gy};
        a1 = (v2f){gx, gy};
        a2 = (v2f){1.0f, 0.0f};
    } else {
        a0 = (v2f){-0.5f * gz * gz, -0.5f * gp * gp};
        a1 = (v2f){gz, gp};
        a2 = (v2f){0.0f, 0.0f};
    }

    const float inv_denom = fast_rcp(sum_p[b] + EPSF);
    const size_t base_idx = (size_t)b * K_PIX;

    float acc[8];
#pragma unroll
    for (int v = 0; v < 8; ++v) acc[v] = 0.0f;

    for (int t = wave; t < NTILES; t += WAVES) {
        v2f b0, b1, b2;
        if (USE_FEAT) {
            const float* tb = feat + ((size_t)b * NTILES + t) * TILE_FLOATS;
            b0 = *(const v2f*)(tb + 0 * 64 + lane * 2);
            b1 = *(const v2f*)(tb + 1 * 64 + lane * 2);
            b2 = *(const v2f*)(tb + 2 * 64 + lane * 2);
            // Unconditional speculative prefetch of the next tile block:
            // worst case overshoots the buffer by ~3KB; invalid-translation
            // prefetches are silently dropped (ISA 10.5), so no guard needed
            // and the loop stays free of EXEC-mask manipulation.
            __builtin_prefetch(tb + (size_t)WAVES * TILE_FLOATS, 0, 1);
        } else {
            const size_t idx = base_idx + (size_t)t * 16 + m;
            float ix, iy, iz, ip, mux, muy, muz, mup, c9;
            pixel_features(idx, inv_denom, p, dxp, dyp, dzp, php, sxp, syp,
                           szp, spp, pcx, pcy, pcz,
                           ix, iy, iz, ip, mux, muy, muz, mup, c9);
            if (h == 0) {
                b0 = (v2f){ix, iy};
                b1 = (v2f){mux * ix, muy * iy};
                b2 = (v2f){c9, 0.0f};
            } else {
                b0 = (v2f){iz, ip};
                b1 = (v2f){muz * iz, mup * ip};
                b2 = (v2f){0.0f, 0.0f};
            }
        }

        v8f c = {0.f, 0.f, 0.f, 0.f, 0.f, 0.f, 0.f, 0.f};
        c = __builtin_amdgcn_wmma_f32_16x16x4_f32(false, a0, false, b0, (short)0, c, false, false);
        c = __builtin_amdgcn_wmma_f32_16x16x4_f32(false, a1, false, b1, (short)0, c, false, false);
        c = __builtin_amdgcn_wmma_f32_16x16x4_f32(false, a2, false, b2, (short)0, c, false, false);

        // c[v] = logit[e = v + 8*h][col m] + SHIFT ; stream the exp-sum
#pragma unroll
        for (int v = 0; v < 8; ++v) acc[v] += __expf(c[v]);
    }

    // Reduce over the 16 columns held within each 16-lane half.
#pragma unroll
    for (int mask = 1; mask <= 8; mask <<= 1) {
#pragma unroll
        for (int v = 0; v < 8; ++v) acc[v] += __shfl_xor(acc[v], mask, 32);
    }
    if (lane == 0) {
#pragma unroll
        for (int v = 0; v < 8; ++v) red[wave][v] = acc[v];       // rows 0..7
    }
    if (lane == 16) {
#pragma unroll
        for (int v = 0; v < 8; ++v) red[wave][8 + v] = acc[v];   // rows 8..15
    }
    __syncthreads();

    if (threadIdx.x < 16) {
        float tot = red[0][threadIdx.x] + red[1][threadIdx.x]
                  + red[2][threadIdx.x] + red[3][threadIdx.x];
        lsebuf[threadIdx.x] = __logf(tot) - SHIFTF;
    }
    __syncthreads();

    if (threadIdx.x == 0) {
        float s = 0.0f;
#pragma unroll
        for (int i = 0; i < 16; ++i) s += lsebuf[i];
        partials[blockIdx.x] = s;
    }
}

// ---------------------------------------------------------------------------
// Kernel 3: deterministic final reduction of the 256 workgroup partials
// ---------------------------------------------------------------------------
__global__ void k_final(const float* __restrict__ partials, float* __restrict__ out) {
    __shared__ float red[256];
    red[threadIdx.x] = partials[threadIdx.x];
    __syncthreads();
    for (int o = 128; o > 0; o >>= 1) {
        if (threadIdx.x < o) red[threadIdx.x] += red[threadIdx.x + o];
        __syncthreads();
    }
    if (threadIdx.x == 0) out[0] = -red[0] / (float)(BATCH * EMIT);
}

// ---------------------------------------------------------------------------
extern "C" void kernel_launch(void* const* d_in, const int* in_sizes, int n_in,
                              void* d_out, int out_size, void* d_ws, size_t ws_size,
                              hipStream_t stream) {
    (void)in_sizes; (void)n_in; (void)out_size;
    const float* p   = (const float*)d_in[0];
    const float* dx  = (const float*)d_in[1];
    const float* dy  = (const float*)d_in[2];
    const float* dz  = (const float*)d_in[3];
    const float* ph  = (const float*)d_in[4];
    const float* sx  = (const float*)d_in[5];
    const float* sy  = (const float*)d_in[6];
    const float* sz  = (const float*)d_in[7];
    const float* sp  = (const float*)d_in[8];
    const float* gt  = (const float*)d_in[9];
    const float* pcx = (const float*)d_in[10];
    const float* pcy = (const float*)d_in[11];
    const float* pcz = (const float*)d_in[12];

    float* ws       = (float*)d_ws;
    float* sum_p    = ws;        // 32 floats
    float* partials = ws + 64;   // 256 floats
    float* feat     = ws + 512;  // fragment buffer (2KB-aligned offset)

    const size_t feat_floats = (size_t)BATCH * NTILES * TILE_FLOATS;  // ~6.3M
    const bool fast = ws_size >= (512 + feat_floats) * sizeof(float);

    k_sum_p<<<BATCH, 256, 0, stream>>>(p, sum_p);

    if (fast) {
        k_features<<<(BATCH * K_PIX) / 256, 256, 0, stream>>>(
            p, dx, dy, dz, ph, sx, sy, sz, sp, pcx, pcy, pcz, sum_p, feat);
        k_mixture<true><<<BATCH * 8, 128, 0, stream>>>(
            p, dx, dy, dz, ph, sx, sy, sz, sp, gt, pcx, pcy, pcz, sum_p,
            feat, partials);
    } else {
        k_mixture<false><<<BATCH * 8, 128, 0, stream>>>(
            p, dx, dy, dz, ph, sx, sy, sz, sp, gt, pcx, pcy, pcz, sum_p,
            nullptr, partials);
    }

    k_final<<<1, 256, 0, stream>>>(partials, (float*)d_out);
}